// DoctoralLoss_19842748908032
// MI455X (gfx1250) — compile-verified
//
#include <hip/hip_runtime.h>
#include <hip/hip_bf16.h>
#include <stdint.h>

// Problem constants (match reference)
#define T_SAMPLES 100
#define B_SIZE    131072
#define C_CLASSES 3

typedef float v2f __attribute__((ext_vector_type(2)));
typedef float v8f __attribute__((ext_vector_type(8)));

// ---------------- deterministic counter-based RNG (graph-replay safe) --------
__device__ __forceinline__ uint32_t hash_u32(uint32_t x) {
    x ^= x >> 16; x *= 0x7feb352du;
    x ^= x >> 15; x *= 0x846ca68bu;
    x ^= x >> 16;
    return x;
}
// uniform in (0,1] from 24 high bits (never 0 -> log() safe)
__device__ __forceinline__ float u01(uint32_t h) {
    return (float)((h >> 8) + 1u) * 5.9604645e-08f; // * 2^-24
}

// ------ exact f32 half-wave sums via V_WMMA_F32_16X16X4_F32 ------------------
// A (16x4 f32): VGPR0 = lane value, VGPR1 = 0  => A[m][0]=v[m], A[m][2]=v[m+16]
// B (4x16 f32): all ones
// D[m][n] = v[m] + v[m+16]  (exact f32 adds in the matrix core)
// Folding D's 8 VGPRs per lane: lanes 0-15 get sum(rows 0..7), lanes 16-31 get
// sum(rows 8..15). Lane 0 + lane 16 together hold the full 32-lane sum — we
// combine them through LDS (cheaper than a ds_bpermute shuffle round-trip).
__device__ __forceinline__ float wave_half_sums(float v) {
#if defined(__gfx1250__) && __has_builtin(__builtin_amdgcn_wmma_f32_16x16x4_f32)
    v2f a; a[0] = v;    a[1] = 0.0f;
    v2f o; o[0] = 1.0f; o[1] = 1.0f;
    v8f c = {};
    c = __builtin_amdgcn_wmma_f32_16x16x4_f32(
        /*neg_a=*/false, a, /*neg_b=*/false, o,
        /*c_mod=*/(short)0, c, /*reuse_a=*/false, /*reuse_b=*/false);
    return ((c[0] + c[1]) + (c[2] + c[3])) + ((c[4] + c[5]) + (c[6] + c[7]));
#else
    for (int off = 1; off < 16; off <<= 1) v += __shfl_xor(v, off, 32);
    return v;
#endif
}

// ---------------- stage 1: per-row Monte-Carlo loss + block partial sums -----
__global__ __launch_bounds__(256)
void doctoral_loss_partials(const float* __restrict__ logits,
                            const float* __restrict__ log_var,
                            const float* __restrict__ p_win,
                            const int*   __restrict__ targets,
                            float*       __restrict__ partials) {
    const int b = blockIdx.x * 256 + threadIdx.x;   // grid exactly covers B: no divergence

    const float l0 = logits[b * 3 + 0];
    const float l1 = logits[b * 3 + 1];
    const float l2 = logits[b * 3 + 2];
    const float lv = log_var[b];
    const float pw = p_win[b];
    const int   tg = targets[b];

    const float stdv = __expf(0.5f * lv);
    const float TWO_PI = 6.28318530718f;

    // ---- aleatoric classification NLL over T Monte-Carlo samples ----
    // Two samples per iteration: 6 normals from exactly 3 Box-Muller pairs
    // (3x log, 3x sqrt, 3x sincos per 2 samples).
    float nll = 0.0f;
    const uint32_t base = (uint32_t)b * (uint32_t)(3 * T_SAMPLES);
    #pragma unroll 2
    for (int i = 0; i < T_SAMPLES / 2; ++i) {
        const uint32_t s = base + 6u * (uint32_t)i;
        const float u1 = u01(hash_u32(s + 0u));
        const float u2 = u01(hash_u32(s + 1u));
        const float u3 = u01(hash_u32(s + 2u));
        const float u4 = u01(hash_u32(s + 3u));
        const float u5 = u01(hash_u32(s + 4u));
        const float u6 = u01(hash_u32(s + 5u));

        const float r1 = sqrtf(-2.0f * __logf(u1));
        const float r2 = sqrtf(-2.0f * __logf(u3));
        const float r3 = sqrtf(-2.0f * __logf(u5));
        float sn1, cs1, sn2, cs2, sn3, cs3;
        __sincosf(TWO_PI * u2, &sn1, &cs1);
        __sincosf(TWO_PI * u4, &sn2, &cs2);
        __sincosf(TWO_PI * u6, &sn3, &cs3);

        // sample A uses (r1*cs1, r1*sn1, r2*cs2); sample B uses (r2*sn2, r3*cs3, r3*sn3)
        {
            const float d0 = fmaf(stdv, r1 * cs1, l0);
            const float d1 = fmaf(stdv, r1 * sn1, l1);
            const float d2 = fmaf(stdv, r2 * cs2, l2);
            const float m   = fmaxf(d0, fmaxf(d1, d2));
            const float lse = m + __logf(__expf(d0 - m) + __expf(d1 - m) + __expf(d2 - m));
            const float dt  = (tg == 0) ? d0 : ((tg == 1) ? d1 : d2);
            nll += lse - dt;
        }
        {
            const float d0 = fmaf(stdv, r2 * sn2, l0);
            const float d1 = fmaf(stdv, r3 * cs3, l1);
            const float d2 = fmaf(stdv, r3 * sn3, l2);
            const float m   = fmaxf(d0, fmaxf(d1, d2));
            const float lse = m + __logf(__expf(d0 - m) + __expf(d1 - m) + __expf(d2 - m));
            const float dt  = (tg == 0) ? d0 : ((tg == 1) ? d1 : d2);
            nll += lse - dt;
        }
    }

    // ---- pinball confidence loss (Q=0.5 => 0.5*|err|; weighted 0.5 => 0.25) ----
    int pred = 0; float best = l0;
    if (l1 > best) { best = l1; pred = 1; }   // first-max tie-break (argmax semantics)
    if (l2 > best) { best = l2; pred = 2; }
    const float correctness = (pred == tg) ? 1.0f : 0.0f;
    const float err = correctness - pw;

    const float invB = 1.0f / (float)B_SIZE;
    const float contrib = nll * (1.0f / ((float)T_SAMPLES * (float)B_SIZE))
                        + 0.25f * fabsf(err) * invB
                        + 0.1f  * __expf(lv) * invB;

    // ---- exact f32 reduction: WMMA per wave, half-sums to LDS, thread0 folds ----
    const float hsum = wave_half_sums(contrib);   // EXEC all-1s here (no divergence yet)

    __shared__ float sdata[16];                   // 8 waves x 2 half-sums
    const int lane = threadIdx.x & 31;
    const int wave = threadIdx.x >> 5;
    if ((lane & 15) == 0) sdata[wave * 2 + (lane >> 4)] = hsum;
    __syncthreads();
    if (threadIdx.x == 0) {
        float s = 0.0f;
        #pragma unroll
        for (int i = 0; i < 16; ++i) s += sdata[i];
        partials[blockIdx.x] = s;
    }
}

// ---------------- stage 2: deterministic fold of 512 partials ----------------
__global__ __launch_bounds__(512)
void doctoral_loss_finalize(const float* __restrict__ partials,
                            float*       __restrict__ out) {
    const float v = partials[threadIdx.x];        // 512 threads, all active
    const float hsum = wave_half_sums(v);

    __shared__ float sdata[32];                   // 16 waves x 2 half-sums
    const int lane = threadIdx.x & 31;
    const int wave = threadIdx.x >> 5;
    if ((lane & 15) == 0) sdata[wave * 2 + (lane >> 4)] = hsum;
    __syncthreads();
    if (threadIdx.x == 0) {
        float s = 0.0f;
        #pragma unroll
        for (int i = 0; i < 32; ++i) s += sdata[i];
        out[0] = s;
    }
}

extern "C" void kernel_launch(void* const* d_in, const int* in_sizes, int n_in,
                              void* d_out, int out_size, void* d_ws, size_t ws_size,
                              hipStream_t stream) {
    const float* logits  = (const float*)d_in[0];   // (B,3)
    const float* log_var = (const float*)d_in[1];   // (B,1)
    const float* p_win   = (const float*)d_in[2];   // (B,1)
    const int*   targets = (const int*)d_in[3];     // (B,)
    float* out      = (float*)d_out;
    float* partials = (float*)d_ws;                 // 512 floats of scratch

    const int nblocks = B_SIZE / 256;               // 512
    doctoral_loss_partials<<<nblocks, 256, 0, stream>>>(logits, log_var, p_win,
                                                        targets, partials);
    doctoral_loss_finalize<<<1, 512, 0, stream>>>(partials, out);
}